// DeformableSincConv1d_57019985822444
// MI455X (gfx1250) — compile-verified
//
#include <hip/hip_runtime.h>
#include <hip/hip_bf16.h>

// ---------------- problem constants (from reference) ----------------
#define SRATE   16000
#define C_OUTN  80
#define KW      51
#define KP      52          // K padded to multiple of 4 (zero filter tap at k=51)
#define STRIDE  10
#define L_IN    32000
#define L_OFF   3195        // (32000-51)/10+1
#define DEF_LEN (L_OFF*KW)  // 162945
#define L_OUT   16290       // (162945-51)/10+1

typedef float v2f __attribute__((ext_vector_type(2)));
typedef float v8f __attribute__((ext_vector_type(8)));

// V_WMMA_F32_16X16X4_F32 : D(16x16 f32) = A(16x4 f32) * B(4x16 f32) + C
// A frag layout: a[j] = A[lane%16][ (lane<16?0:2) + j ]
// B frag layout (mirrors A with N in place of M): b[j] = B[(lane<16?0:2)+j][lane%16]
// C/D layout: c[i] = D[ i + (lane<16?0:8) ][ lane%16 ]
__device__ __forceinline__ v8f wmma4(v2f a, v2f b, v8f c) {
  return __builtin_amdgcn_wmma_f32_16x16x4_f32(false, a, false, b, (short)0, c,
                                               false, false);
}

// ---------------- kernel 1: sinc filter bank (80 x 52, padded) ----------------
__device__ __forceinline__ float lp_term(float f, float t) {
  // 2*f*sinc(2*f*t) with sinc(u)=sin(u)/u ; u==0 -> 2*f
  float u = 2.f * f * t;
  return (u == 0.f) ? (2.f * f) : (__builtin_sinf(u) / t);
}

__global__ __launch_bounds__(256)
void build_filters_kernel(const float* __restrict__ hz,
                          const float* __restrict__ band,
                          float* __restrict__ filt) {
  int i = blockIdx.x * blockDim.x + threadIdx.x;
  if (i >= C_OUTN * KP) return;
  int c = i / KP, k = i % KP;
  float h  = fminf(fmaxf(hz[c],   0.f), 8000.f);
  float bd = fminf(fmaxf(band[c], 3.f), 8000.f);
  float lo = h - 0.5f * bd, hi = h + 0.5f * bd;
  float val;
  if (k >= KW) {
    val = 0.f;                                   // zero pad tap
  } else {
    int d = k - 25; d = d < 0 ? -d : d;          // distance from center
    if (d == 0) {
      val = 1.f;
    } else {
      float t = (float)d / (float)SRATE;
      val = lp_term(hi, t) - lp_term(lo, t);
    }
  }
  filt[c * KP + k] = val / (2.f * bd);
}

// ------------- kernel 2: offset conv (WMMA) + deform gather (lerp) -------------
// block = 256 threads (8 waves) handles one batch x 128 'l' positions.
__global__ __launch_bounds__(256)
void deform_kernel(const float* __restrict__ x,
                   const float* __restrict__ ow,   // (51,1,51)
                   const float* __restrict__ ob,   // (51,)
                   float* __restrict__ deformed) {
  __shared__ float wpad[64 * KP];   // offset_w padded to 64x52 (A matrix)
  __shared__ float xwin[1328];      // x window: 127*10 + 52 = 1322 floats
  __shared__ float bias[64];

  const int tid = threadIdx.x;
  const int b   = blockIdx.y;
  const int l0  = blockIdx.x * 128;
  const int base = l0 * STRIDE;

  for (int i = tid; i < 64 * KP; i += 256) {
    int c = i / KP, j = i % KP;
    wpad[i] = (c < KW && j < KW) ? ow[c * KW + j] : 0.f;
  }
  if (tid < 64) bias[tid] = (tid < KW) ? ob[tid] : 0.f;
  for (int i = tid; i < 1322; i += 256) {
    int g = base + i;
    g = g < L_IN ? g : (L_IN - 1);               // clamp (overhang block)
    xwin[i] = x[b * L_IN + g];
  }
  __syncthreads();

  const int lane = tid & 31;
  const int wv   = tid >> 5;
  const int n    = lane & 15;
  const int kk   = (lane >> 4) << 1;             // 0 or 2
  const int rel  = (wv * 16 + n) * STRIDE;       // window pos of this N column

  v8f acc[4];
  #pragma unroll
  for (int m = 0; m < 4; ++m) acc[m] = (v8f){0.f,0.f,0.f,0.f,0.f,0.f,0.f,0.f};

  #pragma unroll
  for (int ks = 0; ks < 13; ++ks) {
    const int k0 = ks * 4 + kk;
    v2f bb = { xwin[rel + k0], xwin[rel + k0 + 1] };
    #pragma unroll
    for (int m = 0; m < 4; ++m) {
      const int row = (m * 16 + n) * KP + k0;
      v2f aa = { wpad[row], wpad[row + 1] };
      acc[m] = wmma4(aa, bb, acc[m]);
    }
  }

  // epilogue: acc[m][i] = offset for out-channel k = m*16 + i + (lane<16?0:8),
  // position l = l0 + 16*wv + n. Lerp-gather from the same LDS x window.
  const int   l      = l0 + wv * 16 + n;
  const bool  lvalid = (l < L_OFF);
  const float t0f    = (float)(l * STRIDE);
  const int   mrow   = (lane >> 4) * 8;          // 0 or 8
  float* dptr = deformed + b * DEF_LEN + l * KW;

  #pragma unroll
  for (int m = 0; m < 4; ++m) {
    #pragma unroll
    for (int i = 0; i < 8; ++i) {
      const int k = m * 16 + mrow + i;
      if (k < KW && lvalid) {
        float off = acc[m][i] + bias[k];
        float T = t0f + (float)k + off;
        T = fminf(fmaxf(T, t0f), t0f + 50.f);    // clip to [10l, 10l+50]
        int   u  = (int)T;                       // floor (T >= 0)
        float fr = T - (float)u;
        int   r  = u - base;                     // in-window (<= 1320)
        float v  = (1.f - fr) * xwin[r] + fr * xwin[r + 1];
        dptr[k] = v;
      }
    }
  }
}

// ------------- kernel 3: final sinc conv as WMMA GEMM (80 x 16290 per batch) -------------
// block = 256 threads (8 waves) handles one batch x 128 output positions.
__global__ __launch_bounds__(256)
void sincconv_kernel(const float* __restrict__ deformed,
                     const float* __restrict__ filt,   // 80 x 52 padded
                     float* __restrict__ out) {
  __shared__ float flds[C_OUTN * KP];   // 4160 floats
  __shared__ float dwin[1480];          // 127*10 + 202 = 1472 floats

  const int tid  = threadIdx.x;
  const int b    = blockIdx.y;
  const int o0   = blockIdx.x * 128;
  const int base = o0 * STRIDE;

  for (int i = tid; i < C_OUTN * KP; i += 256) flds[i] = filt[i];
  for (int i = tid; i < 1472; i += 256) {
    int g = base + i;
    g = g < DEF_LEN ? g : (DEF_LEN - 1);          // clamp (overhang block)
    dwin[i] = deformed[b * DEF_LEN + g];
  }
  __syncthreads();

  const int lane = tid & 31;
  const int wv   = tid >> 5;
  const int n    = lane & 15;
  const int kk   = (lane >> 4) << 1;              // 0 or 2
  const int rel  = (wv * 16 + n) * STRIDE;

  v8f acc[5];
  #pragma unroll
  for (int m = 0; m < 5; ++m) acc[m] = (v8f){0.f,0.f,0.f,0.f,0.f,0.f,0.f,0.f};

  #pragma unroll
  for (int ks = 0; ks < 13; ++ks) {
    const int k0 = ks * 4 + kk;
    v2f bb = { dwin[rel + k0], dwin[rel + k0 + 1] };
    #pragma unroll
    for (int m = 0; m < 5; ++m) {
      const int row = (m * 16 + n) * KP + k0;
      v2f aa = { flds[row], flds[row + 1] };
      acc[m] = wmma4(aa, bb, acc[m]);
    }
  }

  const int o = o0 + wv * 16 + n;
  if (o < L_OUT) {
    const int mrow = (lane >> 4) * 8;             // 0 or 8
    #pragma unroll
    for (int m = 0; m < 5; ++m) {
      #pragma unroll
      for (int i = 0; i < 8; ++i) {
        const int c = m * 16 + mrow + i;          // < 80 always
        out[((long)b * C_OUTN + c) * L_OUT + o] = acc[m][i];
      }
    }
  }
}

// ---------------------------------- launch ----------------------------------
extern "C" void kernel_launch(void* const* d_in, const int* in_sizes, int n_in,
                              void* d_out, int out_size, void* d_ws, size_t ws_size,
                              hipStream_t stream) {
  const float* x    = (const float*)d_in[0];   // (32, 32000)
  const float* hz   = (const float*)d_in[1];   // (80, 1)
  const float* band = (const float*)d_in[2];   // (80, 1)
  const float* ow   = (const float*)d_in[3];   // (51, 1, 51)
  const float* ob   = (const float*)d_in[4];   // (51,)
  float* out = (float*)d_out;                  // (32, 80, 16290)

  float* filt     = (float*)d_ws;              // 80*52 floats
  float* deformed = (float*)d_ws + 8192;       // 32 * 162945 floats (~20.9 MB)

  build_filters_kernel<<<(C_OUTN * KP + 255) / 256, 256, 0, stream>>>(hz, band, filt);
  deform_kernel<<<dim3(25, 32), 256, 0, stream>>>(x, ow, ob, deformed);
  sincconv_kernel<<<dim3(128, 32), 256, 0, stream>>>(deformed, filt, out);
}